// GIN_13975823581719
// MI455X (gfx1250) — compile-verified
//
#include <hip/hip_runtime.h>

// ---------------------------------------------------------------------------
// GIN network for MI455X (gfx1250, wave32, WMMA).
//   h = act((x + segsum(x[src] -> dst)) @ W + b)  x2   (GIN convs)
//   h = act(h @ W + b)                            x4   (MLP)
//   out = h @ W + b                                    (final, N=64)
// All fp32, V_WMMA_F32_16X16X4_F32 for the GEMMs (fp32-exact matrix path).
//
// GEMM: block = 80 rows (5 M-tiles) x N cols. One wave32 per 16-col N-tile.
// B-panel (128x16) preloaded to 32 VGPR pairs per lane, reused across all 5
// M-tiles -> hot loop is ds_load_b64 + v_wmma only (no VMEM).
// ---------------------------------------------------------------------------

typedef __attribute__((ext_vector_type(2))) float v2f;
typedef __attribute__((ext_vector_type(8))) float v8f;

#define NN_NODES 50000
#define NN_EDGES 600000
#define K_DIM 128            // CH_IN == HID == 128 for every GEMM's K
#define LDS_STRIDE 132       // 132 % 64 = 4 -> conflict-free, 16B/8B aligned
#define M_ROWS 80            // rows per block; 50000 = 625 * 80 exactly
#define LRELU_SLOPE 0.01f

// ---------------------------------------------------------------- zero ------
__global__ void gin_zero_kernel(float* __restrict__ p, int n4) {
    unsigned i = blockIdx.x * blockDim.x + threadIdx.x;
    if (i < (unsigned)n4) ((float4*)p)[i] = make_float4(0.f, 0.f, 0.f, 0.f);
}

// ------------------------------------------------------------- scatter -----
// agg[dst[e]] += x[src[e]]  (segment_sum). One thread per (edge, 4-float
// chunk): coalesced float4 gather, 4 fp32 atomics resolving in L2
// (agg = 25.6 MB << 192 MB L2).
__global__ void gin_scatter_kernel(const float* __restrict__ x,
                                   const int*   __restrict__ src,
                                   const int*   __restrict__ dst,
                                   float*       __restrict__ agg) {
    unsigned t = blockIdx.x * blockDim.x + threadIdx.x;
    const unsigned total = (unsigned)NN_EDGES * (K_DIM / 4);   // 19.2M
    if (t >= total) return;
    const unsigned e = t >> 5;          // 32 chunks per edge
    const unsigned c = t & 31;
    const unsigned s = (unsigned)src[e];
    const unsigned d = (unsigned)dst[e];
    const float4 v = ((const float4*)(x + s * (unsigned)K_DIM))[c];
    float* a = agg + d * (unsigned)K_DIM + c * 4u;
    atomicAdd(a + 0, v.x);
    atomicAdd(a + 1, v.y);
    atomicAdd(a + 2, v.z);
    atomicAdd(a + 3, v.w);
}

// ---------------------------------------------------------------- GEMM -----
// f32 WMMA VGPR layouts (ISA 7.12.2):
//   A 16x4 : lanes 0-15 -> M=lane,    v[0]=K0 v[1]=K1
//            lanes 16-31-> M=lane-16, v[0]=K2 v[1]=K3
//   B 4x16 : lanes 0-15 -> N=lane,    v[0]=K0 v[1]=K1
//            lanes 16-31-> N=lane-16, v[0]=K2 v[1]=K3
//   C/D    : VGPR r: lanes 0-15 -> (M=r, N=lane); lanes 16-31 -> (M=r+8, N=lane-16)
__global__ void gin_gemm_kernel(const float* __restrict__ x,
                                const float* __restrict__ agg,   // may be null
                                const float* __restrict__ W,     // K_DIM x N row-major
                                const float* __restrict__ bias,  // N
                                float*       __restrict__ out,   // M x N
                                int N, int applyAct) {
    __shared__ float sA[M_ROWS * LDS_STRIDE];    // 80*132*4 = 42240 B

    const unsigned rowBase = blockIdx.x * (unsigned)M_ROWS;
    const unsigned uN = (unsigned)N;

    // Stage 80x128 A tile (x + agg) into LDS; GIN's "(1+eps)x + sum" fused.
    for (unsigned c = threadIdx.x; c < M_ROWS * 32u; c += blockDim.x) {
        const unsigned r  = c >> 5;
        const unsigned c4 = (c & 31u) * 4u;
        const unsigned off = (rowBase + r) * (unsigned)K_DIM + c4;
        float4 v = *(const float4*)(x + off);
        if (agg) {
            const float4 g = *(const float4*)(agg + off);
            v.x += g.x; v.y += g.y; v.z += g.z; v.w += g.w;
        }
        *(float4*)(sA + r * LDS_STRIDE + c4) = v;
    }

    const unsigned wave     = threadIdx.x >> 5;
    const unsigned lane     = threadIdx.x & 31u;
    const unsigned halfLane = lane & 15u;
    const bool     hi       = lane >= 16u;
    const unsigned nCol     = wave * 16u + halfLane;
    const float    bval     = bias[nCol];

    // Preload this lane's B column slice (K=128, k%4 in {0,1} or {2,3} by lane half) into 32 VGPR
    // pairs; W is 64 KB, L2-resident, loaded once per block per wave.
    v2f breg[32];
#pragma unroll
    for (int i = 0; i < 32; ++i) {
        const unsigned k0 = 4u * i + (hi ? 2u : 0u);
        breg[i].x = W[k0 * uN + nCol];
        breg[i].y = W[(k0 + 1u) * uN + nCol];
    }

    __syncthreads();

    for (int mt = 0; mt < M_ROWS / 16; ++mt) {   // 5 M-tiles reuse breg
        const float* sRow = sA + (mt * 16u + halfLane) * LDS_STRIDE;
        v8f acc = {};
#pragma unroll
        for (int i = 0; i < 32; ++i) {
            const int k0 = 4 * i + (hi ? 2 : 0);
            v2f a;
            a.x = sRow[k0];
            a.y = sRow[k0 + 1];
            // 8 args: (neg_a, A, neg_b, B, c_mod, C, reuse_a, reuse_b)
            acc = __builtin_amdgcn_wmma_f32_16x16x4_f32(
                false, a, false, breg[i], (short)0, acc, false, false);
        }

        const unsigned rTop = rowBase + mt * 16u + (hi ? 8u : 0u);
#pragma unroll
        for (int r = 0; r < 8; ++r) {
            float h = acc[r] + bval;
            if (applyAct) h = fmaxf(h, LRELU_SLOPE * h);   // leaky relu, slope>0
            out[(rTop + r) * uN + nCol] = h;
        }
    }
}

// --------------------------------------------------------------- launch ----
extern "C" void kernel_launch(void* const* d_in, const int* in_sizes, int n_in,
                              void* d_out, int out_size, void* d_ws, size_t ws_size,
                              hipStream_t stream) {
    const float* feats = (const float*)d_in[0];
    const int*   src   = (const int*)  d_in[1];   // jnp.int64 -> int32 (x64 off)
    const int*   dst   = (const int*)  d_in[2];
    const float* W1  = (const float*)d_in[3];
    const float* b1  = (const float*)d_in[4];
    const float* W2  = (const float*)d_in[5];
    const float* b2  = (const float*)d_in[6];
    const float* Wl1 = (const float*)d_in[7];
    const float* bl1 = (const float*)d_in[8];
    const float* Wl2 = (const float*)d_in[9];
    const float* bl2 = (const float*)d_in[10];
    const float* Wl3 = (const float*)d_in[11];
    const float* bl3 = (const float*)d_in[12];
    const float* Wl4 = (const float*)d_in[13];
    const float* bl4 = (const float*)d_in[14];
    const float* Wl5 = (const float*)d_in[15];
    const float* bl5 = (const float*)d_in[16];
    float* out = (float*)d_out;

    const size_t nodeElems = (size_t)NN_NODES * K_DIM;   // 6.4M floats
    float* agg = (float*)d_ws;
    float* h0  = agg + nodeElems;
    float* h1  = h0 + nodeElems;

    const int  zero4 = (int)(nodeElems / 4);
    const dim3 zGrid((zero4 + 255) / 256), zBlk(256);
    const unsigned scatterT = (unsigned)NN_EDGES * 32u;
    const dim3 sGrid((scatterT + 255) / 256), sBlk(256);
    const dim3 gGrid(NN_NODES / M_ROWS);                 // 625, exact
    const dim3 gBlk128(256);                             // 8 waves (N=128)
    const dim3 gBlk64(128);                              // 4 waves (N=64)

    // --- GIN conv 1: agg = scatter(feats); h0 = act((feats+agg)@W1+b1)
    gin_zero_kernel<<<zGrid, zBlk, 0, stream>>>(agg, zero4);
    gin_scatter_kernel<<<sGrid, sBlk, 0, stream>>>(feats, src, dst, agg);
    gin_gemm_kernel<<<gGrid, gBlk128, 0, stream>>>(feats, agg, W1, b1, h0, 128, 1);

    // --- GIN conv 2
    gin_zero_kernel<<<zGrid, zBlk, 0, stream>>>(agg, zero4);
    gin_scatter_kernel<<<sGrid, sBlk, 0, stream>>>(h0, src, dst, agg);
    gin_gemm_kernel<<<gGrid, gBlk128, 0, stream>>>(h0, agg, W2, b2, h1, 128, 1);

    // --- MLP stack
    gin_gemm_kernel<<<gGrid, gBlk128, 0, stream>>>(h1, nullptr, Wl1, bl1, h0, 128, 1);
    gin_gemm_kernel<<<gGrid, gBlk128, 0, stream>>>(h0, nullptr, Wl2, bl2, h1, 128, 1);
    gin_gemm_kernel<<<gGrid, gBlk128, 0, stream>>>(h1, nullptr, Wl3, bl3, h0, 128, 1);
    gin_gemm_kernel<<<gGrid, gBlk128, 0, stream>>>(h0, nullptr, Wl4, bl4, h1, 128, 1);

    // --- final projection, N=64, no activation
    gin_gemm_kernel<<<gGrid, gBlk64, 0, stream>>>(h1, nullptr, Wl5, bl5, out, 64, 0);
}